// ECELoss_45492293599340
// MI455X (gfx1250) — compile-verified
//
#include <hip/hip_runtime.h>
#include <stdint.h>

#define N_BINS 10
#define C_DIM 128            // classes (fixed by reference)
#define ROWS_PER_TILE 8      // rows staged per TDM descriptor (4 KB)
#define WAVES_PER_BLOCK 4
#define BLOCK_THREADS 128

typedef __attribute__((ext_vector_type(4))) unsigned int u32x4;
typedef __attribute__((ext_vector_type(8))) int i32x8;
typedef __attribute__((ext_vector_type(4))) int i32x4;

#if __has_builtin(__builtin_amdgcn_tensor_load_to_lds) && \
    __has_builtin(__builtin_amdgcn_s_wait_tensorcnt)
#define USE_TDM 1
#pragma message("CDNA5: TDM path enabled (tensor_load_to_lds + s_wait_tensorcnt)")
#else
#define USE_TDM 0
#pragma message("CDNA5: TDM builtins NOT found - falling back to global_load_b128")
#endif

// ------------------------------------------------- per-row wave32 reduction
// v = this lane's 4 contiguous logits of one row (lane*4 .. lane*4+3).
__device__ __forceinline__ void ece_row_reduce(float4 v, int lane, int label,
                                               float* s_bins) {
  int base = lane << 2;
  float m = v.x; int a = base;
  if (v.y > m) { m = v.y; a = base + 1; }
  if (v.z > m) { m = v.z; a = base + 2; }
  if (v.w > m) { m = v.w; a = base + 3; }
#pragma unroll
  for (int off = 16; off > 0; off >>= 1) {   // max with first-index tiebreak
    float om = __shfl_xor(m, off, 32);
    int   oa = __shfl_xor(a, off, 32);
    if (om > m || (om == m && oa < a)) { m = om; a = oa; }
  }
  float se = expf(v.x - m) + expf(v.y - m) + expf(v.z - m) + expf(v.w - m);
#pragma unroll
  for (int off = 16; off > 0; off >>= 1) se += __shfl_xor(se, off, 32);
  if (lane == 0) {
    float conf = 1.0f / se;                           // max softmax prob
    int bin = (int)ceilf(conf * (float)N_BINS) - 1;   // (i/n,(i+1)/n] bins
    bin = bin < 0 ? 0 : (bin > N_BINS - 1 ? N_BINS - 1 : bin);
    atomicAdd(&s_bins[bin], 1.0f);                            // count
    atomicAdd(&s_bins[N_BINS + bin], conf);                   // conf sum
    atomicAdd(&s_bins[2 * N_BINS + bin], (a == label) ? 1.0f : 0.0f); // acc
  }
}

#if USE_TDM
// Issue one TDM descriptor: copy `rows` x 128 fp32 tile (row-major,
// stride 128) from global `gsrc` into LDS byte address `lds_addr`.
// D# layout per CDNA5 ISA ch.8 (2D tensor: groups 2/3 zero).
__device__ __forceinline__ void tdm_load_tile(const float* gsrc,
                                              unsigned int lds_addr,
                                              int rows) {
  uint64_t ga = (uint64_t)(uintptr_t)gsrc;
  u32x4 g0;
  g0.x = 0x1u;                                   // count=1, user descriptor
  g0.y = lds_addr;                               // lds_addr [63:32]
  g0.z = (unsigned int)ga;                       // global_addr[31:0]
  g0.w = (unsigned int)((ga >> 32) & 0x1FFFFFFu) // global_addr[56:32]
         | (2u << 30);                           // type=2 ("image")
  i32x8 g1;
  g1[0] = (2 << 16);                  // workgroup_mask=0, data_size=2 (4B)
  g1[1] = (C_DIM << 16);              // tensor_dim0 = 128 (low16 @ bits 63:48)
  g1[2] = ((rows & 0xFFFF) << 16);    // tensor_dim0 hi=0; tensor_dim1 = rows
  g1[3] = (C_DIM << 16);              // tensor_dim1 hi=0; tile_dim0 = 128
  g1[4] = (rows & 0xFFFF);            // tile_dim1 = rows; tile_dim2 = 0
  g1[5] = C_DIM;                      // tensor_dim0_stride = 128 (low32)
  g1[6] = 0;                          // stride0 hi, stride1 low (unused 2D)
  g1[7] = 0;
  i32x4 z4 = {0, 0, 0, 0};
#if __clang_major__ >= 23
  i32x8 z8 = {0, 0, 0, 0, 0, 0, 0, 0};
  __builtin_amdgcn_tensor_load_to_lds(g0, g1, z4, z4, z8, 0);
#else
  __builtin_amdgcn_tensor_load_to_lds(g0, g1, z4, z4, 0);
#endif
}
#endif

// ---------------------------------------------------------- pass 1 (first
// function in file so the returned asm snippet shows the TDM sequence)
__global__ void __launch_bounds__(BLOCK_THREADS)
ece_pass1(const float* __restrict__ logits, const int* __restrict__ labels,
          float* __restrict__ bins, int N) {
  __shared__ float s_bins[3 * N_BINS];
#if USE_TDM
  __shared__ __align__(16) float stage[WAVES_PER_BLOCK][2][ROWS_PER_TILE * C_DIM];
#endif
  if (threadIdx.x < 3 * N_BINS) s_bins[threadIdx.x] = 0.0f;  // 30 < 128
  __syncthreads();

  const int lane  = threadIdx.x & 31;
  const int wave  = __builtin_amdgcn_readfirstlane(threadIdx.x >> 5);
  const int gwave = __builtin_amdgcn_readfirstlane(
      (int)(blockIdx.x * WAVES_PER_BLOCK) + wave);
  const int nwaves = (int)gridDim.x * WAVES_PER_BLOCK;

#if USE_TDM
  const long long tileStride = (long long)nwaves * ROWS_PER_TILE;
  long long r0 = (long long)gwave * ROWS_PER_TILE;
  unsigned int lds0 = (unsigned int)(uintptr_t)
      (__attribute__((address_space(3))) float*)&stage[wave][0][0];
  unsigned int lds1 = (unsigned int)(uintptr_t)
      (__attribute__((address_space(3))) float*)&stage[wave][1][0];
  int buf = 0;
  if (r0 < (long long)N) {                      // prime the pipeline
    long long rem = (long long)N - r0;
    int rows = rem < ROWS_PER_TILE ? (int)rem : ROWS_PER_TILE;
    tdm_load_tile(logits + r0 * C_DIM, lds0, rows);
  }
  for (long long r = r0; r < (long long)N; r += tileStride) {
    long long rem = (long long)N - r;
    int rows = rem < ROWS_PER_TILE ? (int)rem : ROWS_PER_TILE;
    long long rn = r + tileStride;
    if (rn < (long long)N) {                    // prefetch next tile (async)
      long long remn = (long long)N - rn;
      int nrows = remn < ROWS_PER_TILE ? (int)remn : ROWS_PER_TILE;
      tdm_load_tile(logits + rn * C_DIM, buf ? lds0 : lds1, nrows);
      __builtin_amdgcn_s_wait_tensorcnt(1);     // current tile has landed
    } else {
      __builtin_amdgcn_s_wait_tensorcnt(0);
    }
    asm volatile("" ::: "memory");
    const float* tile = &stage[wave][buf][0];
    for (int j = 0; j < rows; ++j) {
      float4 v = *reinterpret_cast<const float4*>(tile + j * C_DIM + (lane << 2));
      ece_row_reduce(v, lane, labels[r + j], s_bins);
    }
    asm volatile("" ::: "memory");
    buf ^= 1;
  }
#else
  // Fallback: direct streaming global_load_b128, one wave per row.
  for (long long row = gwave; row < (long long)N; row += nwaves) {
    const float4* rp = reinterpret_cast<const float4*>(logits + row * C_DIM);
    float4 v = rp[lane];
    ece_row_reduce(v, lane, labels[row], s_bins);
  }
#endif

  __syncthreads();
  if (threadIdx.x < 3 * N_BINS) {
    float s = s_bins[threadIdx.x];
    if (s != 0.0f) atomicAdd(&bins[threadIdx.x], s);
  }
}

// ---------------------------------------------------------------- zero bins
__global__ void ece_zero_bins(float* bins) {
  int i = threadIdx.x;
  if (i < 3 * N_BINS) bins[i] = 0.0f;
}

// ----------------------------------------------------------------- finalize
__global__ void ece_finalize(const float* __restrict__ bins,
                             float* __restrict__ out, int N) {
  if (threadIdx.x != 0 || blockIdx.x != 0) return;
  float ece = 0.0f, oe = 0.0f;
  float invN = 1.0f / (float)N;
  for (int i = 0; i < N_BINS; ++i) {
    float cnt = bins[i];
    float cs  = bins[N_BINS + i];
    float as  = bins[2 * N_BINS + i];
    float prop = cnt * invN;
    bool nonempty = cnt > 0.0f;
    float denom = fmaxf(cnt, 1.0f);
    float aib = nonempty ? as / denom : 0.0f;
    float cib = nonempty ? cs / denom : 0.0f;
    float CE  = cib - aib;
    float aCE = nonempty ? fabsf(CE) : 0.0f;
    ece += aCE * prop;
    oe  += (nonempty ? cib * fmaxf(CE, 0.0f) : 0.0f) * prop;
    out[1 + i]  = aib;   // acc_in_bin
    out[12 + i] = prop;  // prop_in_bin
    out[22 + i] = aCE;   // abs_CE
  }
  out[0]  = ece;
  out[11] = oe;
}

// -------------------------------------------------------------------- launch
extern "C" void kernel_launch(void* const* d_in, const int* in_sizes, int n_in,
                              void* d_out, int out_size, void* d_ws,
                              size_t ws_size, hipStream_t stream) {
  const float* logits = (const float*)d_in[0];
  const int*   labels = (const int*)d_in[1];
  int N = in_sizes[1];               // one label per row
  float* bins = (float*)d_ws;        // [3 * N_BINS] accumulators

  ece_zero_bins<<<1, 32, 0, stream>>>(bins);

  long long tiles = ((long long)N + ROWS_PER_TILE - 1) / ROWS_PER_TILE;
  long long blk = (tiles + WAVES_PER_BLOCK - 1) / WAVES_PER_BLOCK;
  int blocks = blk > 2048 ? 2048 : (blk < 1 ? 1 : (int)blk);

  ece_pass1<<<blocks, BLOCK_THREADS, 0, stream>>>(logits, labels, bins, N);
  ece_finalize<<<1, 32, 0, stream>>>(bins, (float*)d_out, N);
}